// GraphLayer_45999099740494
// MI455X (gfx1250) — compile-verified
//
#include <hip/hip_runtime.h>

#define NN   80000
#define NE   1280000
#define CH   64
#define NC   (NN * CH)
#define LN_EPS 1e-5f

typedef float v2f __attribute__((ext_vector_type(2)));
typedef float v8f __attribute__((ext_vector_type(8)));

// ---------------------------------------------------------------- utilities
__global__ void zero_f32(float* __restrict__ p, int n) {
    int i = blockIdx.x * blockDim.x + threadIdx.x;
    if (i < n) p[i] = 0.f;
}

// deg[dst] += w  (edge-weight in-degree)
__global__ void deg_accum(const int* __restrict__ dst, const float* __restrict__ w,
                          float* __restrict__ deg) {
    int e = blockIdx.x * blockDim.x + threadIdx.x;
    if (e < NE) atomicAdd(&deg[dst[e]], w[e]);
}

// deg -> deg^{-1/2} in place (0 where deg==0)
__global__ void deg_to_dis(float* __restrict__ deg) {
    int i = blockIdx.x * blockDim.x + threadIdx.x;
    if (i < NN) {
        float d = deg[i];
        deg[i] = (d > 0.f) ? rsqrtf(fmaxf(d, 1e-12f)) : 0.f;
    }
}

// norm[e] = dis[src]*w*dis[dst]  (layer-invariant, computed once)
__global__ void edge_norm(const int* __restrict__ src, const int* __restrict__ dst,
                          const float* __restrict__ w, const float* __restrict__ dis,
                          float* __restrict__ nrm) {
    int e = blockIdx.x * blockDim.x + threadIdx.x;
    if (e < NE) nrm[e] = dis[src[e]] * w[e] * dis[dst[e]];
}

// ---------------------------------------------------------------- fp32 WMMA GEMM
// H[N,64] = X[N,64] @ W[64,64], one wave per 16x16 output tile,
// K marched in steps of 4 via V_WMMA_F32_16X16X4_F32.
__global__ __launch_bounds__(256) void gemm_wmma(const float* __restrict__ X,
                                                 const float* __restrict__ W,
                                                 float* __restrict__ H) {
    const int wid   = threadIdx.x >> 5;
    const int lane  = threadIdx.x & 31;
    const int tile  = blockIdx.x * (blockDim.x >> 5) + wid;   // 20000 tiles exactly
    const int m0    = (tile >> 2) << 4;      // 5000 row tiles
    const int n0    = (tile & 3) << 4;       // 4 col tiles
    const int half  = lane >> 4;             // 0 | 1
    const int l16   = lane & 15;

    v8f acc = {};
    const float* xrow = X + (size_t)(m0 + l16) * CH;
#pragma unroll
    for (int k0 = 0; k0 < CH; k0 += 4) {
        // A fragment: lane<16 -> K=k0,k0+1 ; lane>=16 -> K=k0+2,k0+3 (contiguous pair)
        v2f a = *(const v2f*)(xrow + k0 + 2 * half);
        // B fragment: VGPR p, lane-half h -> row K=k0+2h+p, col N=n0+l16
        v2f b;
        b.x = W[(size_t)(k0 + 2 * half + 0) * CH + n0 + l16];
        b.y = W[(size_t)(k0 + 2 * half + 1) * CH + n0 + l16];
        acc = __builtin_amdgcn_wmma_f32_16x16x4_f32(false, a, false, b,
                                                    (short)0, acc, false, false);
    }
    // C/D: VGPR r, lane-half h -> row M = m0 + r + 8h, col N = n0 + l16
    float* hp = H + n0 + l16;
#pragma unroll
    for (int r = 0; r < 8; ++r)
        hp[(size_t)(m0 + r + 8 * half) * CH] = acc[r];
}

// ---------------------------------------------------------------- agg = x + bias (skip + conv bias)
__global__ void init_agg(const float* __restrict__ x, const float* __restrict__ b,
                         float* __restrict__ agg) {
    int i = blockIdx.x * blockDim.x + threadIdx.x;      // over NC/4 float4s
    if (i < NC / 4) {
        const float4 xv = ((const float4*)x)[i];
        const float4 bv = ((const float4*)b)[i & 15];   // CH/4 == 16
        float4 r;
        r.x = xv.x + bv.x; r.y = xv.y + bv.y; r.z = xv.z + bv.z; r.w = xv.w + bv.w;
        ((float4*)agg)[i] = r;
    }
}

// ---------------------------------------------------------------- edge scatter: agg[dst] += norm*h[src]
// 16 threads/edge, float4 gather from h (L2-resident), 4 f32 atomics each.
__global__ void scatter_edges(const int* __restrict__ src, const int* __restrict__ dst,
                              const float* __restrict__ nrm, const float* __restrict__ H,
                              float* __restrict__ agg) {
    int t = blockIdx.x * blockDim.x + threadIdx.x;
    if (t >= NE * 16) return;
    const int e  = t >> 4;
    const int c4 = (t & 15) << 2;
    const int s  = src[e];
    const int d  = dst[e];
    const float nm = nrm[e];
    const float4 hv = *(const float4*)(H + (size_t)s * CH + c4);
    float* ap = agg + (size_t)d * CH + c4;
    atomicAdd(ap + 0, nm * hv.x);
    atomicAdd(ap + 1, nm * hv.y);
    atomicAdd(ap + 2, nm * hv.z);
    atomicAdd(ap + 3, nm * hv.w);
}

// ---------------------------------------------------------------- graph-LN stats over relu(agg)
__global__ __launch_bounds__(256) void relu_stats(const float* __restrict__ agg,
                                                  float* __restrict__ stats) {
    __shared__ float ss[256], sq[256];
    float s = 0.f, s2 = 0.f;
    for (int i = blockIdx.x * blockDim.x + threadIdx.x; i < NC;
         i += gridDim.x * blockDim.x) {
        float v = fmaxf(agg[i], 0.f);
        s += v; s2 += v * v;
    }
    ss[threadIdx.x] = s; sq[threadIdx.x] = s2;
    __syncthreads();
    for (int o = 128; o > 0; o >>= 1) {
        if (threadIdx.x < o) { ss[threadIdx.x] += ss[threadIdx.x + o];
                               sq[threadIdx.x] += sq[threadIdx.x + o]; }
        __syncthreads();
    }
    if (threadIdx.x == 0) { atomicAdd(&stats[0], ss[0]); atomicAdd(&stats[1], sq[0]); }
}

__global__ void ln_finalize(float* __restrict__ stats) {
    const float inv_n = 1.f / (float)NC;
    float m   = stats[0] * inv_n;
    float var = stats[1] * inv_n - m * m;
    float sd  = sqrtf(fmaxf(var, 0.f));
    stats[2]  = m;
    stats[3]  = 1.f / (sd + LN_EPS);
}

// out = (relu(agg) - mean)*inv_std * ln_w[c] + ln_b[c]
__global__ void relu_ln_apply(const float* __restrict__ agg, const float* __restrict__ stats,
                              const float* __restrict__ lw, const float* __restrict__ lb,
                              float* __restrict__ out) {
    int i = blockIdx.x * blockDim.x + threadIdx.x;
    if (i < NC) {
        const float m   = stats[2];
        const float inv = stats[3];
        const int   c   = i & (CH - 1);
        float v = fmaxf(agg[i], 0.f);
        out[i] = (v - m) * inv * lw[c] + lb[c];
    }
}

// ---------------------------------------------------------------- host driver
extern "C" void kernel_launch(void* const* d_in, const int* in_sizes, int n_in,
                              void* d_out, int out_size, void* d_ws, size_t ws_size,
                              hipStream_t stream) {
    const float* x    = (const float*)d_in[0];
    const int*   eidx = (const int*)d_in[1];
    const float* ew   = (const float*)d_in[2];
    const float* Wl[3] = { (const float*)d_in[3], (const float*)d_in[5], (const float*)d_in[7] };
    const float* bl[3] = { (const float*)d_in[4], (const float*)d_in[6], (const float*)d_in[8] };
    const float* lnw[2] = { (const float*)d_in[9],  (const float*)d_in[11] };
    const float* lnb[2] = { (const float*)d_in[10], (const float*)d_in[12] };

    const int* src = eidx;        // edge_index[0]
    const int* dst = eidx + NE;   // edge_index[1]

    float* ws   = (float*)d_ws;
    float* dis  = ws;                 // N
    float* nrm  = dis + NN;           // E
    float* buf0 = nrm + NE;           // N*C
    float* buf1 = buf0 + NC;          // N*C
    float* buf2 = buf1 + NC;          // N*C
    float* stats = buf2 + NC;         // 8

    const int TB = 256;
    const int gN  = (NN + TB - 1) / TB;
    const int gE  = (NE + TB - 1) / TB;
    const int gNC = (NC + TB - 1) / TB;
    const int gNC4 = (NC / 4 + TB - 1) / TB;
    const int gSc = (NE * 16 + TB - 1) / TB;
    const int gGemm = (NN / 16) * (CH / 16) / 8;   // 8 waves/block, one tile/wave

    // --- graph normalization coefficients (layer-invariant) ---
    zero_f32<<<gN, TB, 0, stream>>>(dis, NN);
    deg_accum<<<gE, TB, 0, stream>>>(dst, ew, dis);
    deg_to_dis<<<gN, TB, 0, stream>>>(dis);
    edge_norm<<<gE, TB, 0, stream>>>(src, dst, ew, dis, nrm);

    // buffer rotation: (X, H, AGG, Xnext)
    const float* Xc[3]   = { x,    buf2, buf0 };
    float*       Hb[3]   = { buf0, buf0, buf2 };
    float*       Ab[3]   = { buf1, buf1, buf1 };
    float*       Xn[2]   = { buf2, buf0 };

    for (int l = 0; l < 3; ++l) {
        gemm_wmma<<<gGemm, TB, 0, stream>>>(Xc[l], Wl[l], Hb[l]);
        init_agg<<<gNC4, TB, 0, stream>>>(Xc[l], bl[l], Ab[l]);
        scatter_edges<<<gSc, TB, 0, stream>>>(src, dst, nrm, Hb[l], Ab[l]);
        if (l < 2) {
            zero_f32<<<1, TB, 0, stream>>>(stats, 4);
            relu_stats<<<1024, TB, 0, stream>>>(Ab[l], stats);
            ln_finalize<<<1, 1, 0, stream>>>(stats);
            relu_ln_apply<<<gNC, TB, 0, stream>>>(Ab[l], stats, lnw[l], lnb[l], Xn[l]);
        }
    }

    // tuple output: (x, edge_index, edge_attr) flat
    hipMemcpyAsync(d_out, Ab[2], (size_t)NC * sizeof(float),
                   hipMemcpyDeviceToDevice, stream);
    hipMemcpyAsync((char*)d_out + (size_t)NC * 4, eidx, (size_t)2 * NE * 4,
                   hipMemcpyDeviceToDevice, stream);
    hipMemcpyAsync((char*)d_out + ((size_t)NC + 2 * NE) * 4, ew, (size_t)NE * 4,
                   hipMemcpyDeviceToDevice, stream);
}